// FISF_87050397155461
// MI455X (gfx1250) — compile-verified
//
#include <hip/hip_runtime.h>
#include <hip/hip_bf16.h>
#include <math.h>

#define FEAT 128
#define KLOW 12          // int(128 * 0.1)
#define MAX_HOPS_C 16
#define NUM_ITER_C 20
#define BIGD 1000000000
#define ALPHA_C 0.9f
#define BETA_C 0.85f

typedef float v2f __attribute__((ext_vector_type(2)));
typedef float v8f __attribute__((ext_vector_type(8)));

static __device__ __forceinline__ unsigned int splitmix32(unsigned int x) {
  x += 0x9e3779b9u;
  x ^= x >> 16; x *= 0x21f0aaadu;
  x ^= x >> 15; x *= 0x735a2d97u;
  x ^= x >> 15;
  return x;
}

// ---------------- utility ----------------
__global__ void k_zero(float* p, long long n) {
  long long i = (long long)blockIdx.x * blockDim.x + threadIdx.x;
  if (i < n) p[i] = 0.0f;
}

// ---------------- BFS ----------------
__global__ void k_bfs_init_maskcol(int* dist, const unsigned char* mask, int colf, int n) {
  int v = blockIdx.x * blockDim.x + threadIdx.x;
  if (v < n) dist[v] = mask[(size_t)v * FEAT + colf] ? 0 : BIGD;
}

__global__ void k_bfs_init_precol(int* dist, const unsigned char* mask, const int* preF, int n) {
  int v = blockIdx.x * blockDim.x + threadIdx.x;
  if (v < n) dist[v] = mask[(size_t)v * FEAT + preF[0]] ? 0 : BIGD;
}

__global__ void k_bfs_relax(int* __restrict__ dnext, const int* __restrict__ dcur,
                            const int* __restrict__ row, const int* __restrict__ col, int e) {
  int i = blockIdx.x * blockDim.x + threadIdx.x;
  if (i < e) atomicMin(&dnext[row[i]], dcur[col[i]] + 1);
}

__global__ void k_bfs_fin(float* fd, const int* d, int n) {
  int v = blockIdx.x * blockDim.x + threadIdx.x;
  if (v < n) { int dv = d[v]; fd[v] = (dv >= BIGD) ? 0.0f : (float)dv; }
}

// batched (12-way) BFS for the low-feature seeds
__global__ void k_bfs12_init(int* d, const int* rn, int n) {
  int idx = blockIdx.x * blockDim.x + threadIdx.x;
  if (idx < KLOW * n) {
    int b = idx / n; int v = idx - b * n;
    d[idx] = (v == rn[b]) ? 0 : BIGD;
  }
}

__global__ void k_bfs12_relax(int* __restrict__ dn, const int* __restrict__ dc,
                              const int* __restrict__ row, const int* __restrict__ col,
                              int e, int n) {
  long long idx = (long long)blockIdx.x * blockDim.x + threadIdx.x;
  if (idx < (long long)KLOW * e) {
    int b = (int)(idx / e); int ei = (int)(idx - (long long)b * e);
    int base = b * n;
    atomicMin(&dn[base + row[ei]], dc[base + col[ei]] + 1);
  }
}

// ---------------- adjacency build ----------------
__global__ void k_edge_w(float* __restrict__ w, float* __restrict__ deg,
                         const float* __restrict__ fd,
                         const int* __restrict__ row, const int* __restrict__ col, int e) {
  int i = blockIdx.x * blockDim.x + threadIdx.x;
  if (i < e) {
    float diff = fd[col[i]] - fd[row[i]] + 1.0f;
    float wi = powf(ALPHA_C, diff);
    w[i] = wi;
    atomicAdd(&deg[row[i]], wi);
  }
}

__global__ void k_norm_edges(float* __restrict__ w, const float* __restrict__ deg,
                             const int* __restrict__ row, int e) {
  int i = blockIdx.x * blockDim.x + threadIdx.x;
  if (i < e) { float d = deg[row[i]]; w[i] = (d == 0.0f) ? 0.0f : w[i] * (1.0f / d); }
}

__global__ void k_pc(float* __restrict__ pc, const int* __restrict__ d12,
                     const float* __restrict__ fn2d2, int n) {
  int idx = blockIdx.x * blockDim.x + threadIdx.x;
  if (idx < KLOW * n) {
    int b = idx / n; int v = idx - b * n;
    int dv = d12[idx];
    float fm = (dv >= BIGD) ? 0.0f : (float)dv;
    pc[idx] = powf(ALPHA_C, fn2d2[v]) * powf(BETA_C, fm);
  }
}

__global__ void k_deglow(float* __restrict__ deg, const float* __restrict__ pc,
                         const int* __restrict__ row, const int* __restrict__ col,
                         int e, int n) {
  long long idx = (long long)blockIdx.x * blockDim.x + threadIdx.x;
  if (idx < (long long)KLOW * e) {
    int b = (int)(idx / e); int ei = (int)(idx - (long long)b * e);
    int base = b * n;
    float w = ALPHA_C * pc[base + col[ei]] / pc[base + row[ei]];
    atomicAdd(&deg[base + row[ei]], w);
  }
}

__global__ void k_inv(float* p, long long m) {
  long long i = (long long)blockIdx.x * blockDim.x + threadIdx.x;
  if (i < m) { float d = p[i]; p[i] = (d == 0.0f) ? 0.0f : 1.0f / d; }
}

// ---------------- propagation ----------------
// Wave-level grid-stride over edges; each lane moves a float4 (128 feats =
// 32 lanes * 4). The next edge's source row is prefetched (global_prefetch_b8)
// while the current gather/atomic-scatter is in flight; the whole working set
// is L2-resident on MI455X (192 MB), so this hides L2 latency.
__global__ void k_spmm(float* __restrict__ acc, const float* __restrict__ cur,
                       const float* __restrict__ a, const int* __restrict__ row,
                       const int* __restrict__ col, int e) {
  int lane = threadIdx.x & 31;
  int wid = (int)(((long long)blockIdx.x * blockDim.x + threadIdx.x) >> 5);
  int stride = (int)(((long long)gridDim.x * blockDim.x) >> 5);
  for (int i = wid; i < e; i += stride) {
    int r = row[i], c = col[i];
    float av = a[i];
    int inext = i + stride;
    if (inext < e) {
      int cn = col[inext];
      __builtin_prefetch(cur + (size_t)cn * FEAT + lane * 4, 0, 3);
    }
    float4 v = *(const float4*)(cur + (size_t)c * FEAT + lane * 4);
    float* dst = acc + (size_t)r * FEAT + lane * 4;
    atomicAdd(dst + 0, av * v.x);
    atomicAdd(dst + 1, av * v.y);
    atomicAdd(dst + 2, av * v.z);
    atomicAdd(dst + 3, av * v.w);
  }
}

// phase-2: per-column coefficient (high: shared a_high[e]; low: recomputed from pc/invdl)
__global__ void k_spmm2(float* __restrict__ acc, const float* __restrict__ cur,
                        const float* __restrict__ ah, const float* __restrict__ pc,
                        const float* __restrict__ invdl, const int* __restrict__ lowRank,
                        const int* __restrict__ row, const int* __restrict__ col,
                        int e, int n) {
  __shared__ int slr[FEAT];
  if (threadIdx.x < FEAT) slr[threadIdx.x] = lowRank[threadIdx.x];
  __syncthreads();
  int lane = threadIdx.x & 31;
  int wid = (int)(((long long)blockIdx.x * blockDim.x + threadIdx.x) >> 5);
  int stride = (int)(((long long)gridDim.x * blockDim.x) >> 5);
  for (int i = wid; i < e; i += stride) {
    int r = row[i], c = col[i];
    float a_h = ah[i];
    int inext = i + stride;
    if (inext < e) {
      int cn = col[inext];
      __builtin_prefetch(cur + (size_t)cn * FEAT + lane * 4, 0, 3);
    }
    float4 v = *(const float4*)(cur + (size_t)c * FEAT + lane * 4);
    float vv[4] = {v.x, v.y, v.z, v.w};
    float* dst = acc + (size_t)r * FEAT + lane * 4;
#pragma unroll
    for (int j = 0; j < 4; ++j) {
      int f = lane * 4 + j;
      int lr = slr[f];
      float cf;
      if (lr < 0) {
        cf = a_h;
      } else {
        size_t base = (size_t)lr * n;
        cf = ALPHA_C * pc[base + c] / pc[base + r] * invdl[base + r];
      }
      atomicAdd(dst + j, cf * vv[j]);
    }
  }
}

__global__ void k_apply(float* __restrict__ o, const unsigned char* __restrict__ m,
                        const float* __restrict__ x, const float* __restrict__ acc,
                        long long nf) {
  long long i = (long long)blockIdx.x * blockDim.x + threadIdx.x;
  if (i < nf) o[i] = m[i] ? x[i] : acc[i];
}

// ---------------- variance via V_WMMA_F32_16X16X4_F32 ----------------
// One wave per 16-feature group. A = ones(16x4); B = 4x16 tile of `out`.
// D accumulates column sums (replicated over M); row M=0 lives in accumulator
// element 0 on lanes 0..15. Main loop covers n&~7 rows with UNCONDITIONAL
// loads (no exec-mask churn) and two independent accumulator pairs so the
// wmma D->C chains interleave; the <=7 row tail is folded in scalar by lanes
// 0..15 after the loop.
__global__ void k_var_wmma(const float* __restrict__ out, float* __restrict__ var, int n) {
  int lane = threadIdx.x & 31;
  int f0 = blockIdx.x * 16;
  int nIdx = lane & 15;
  int khalf = lane >> 4;
  v2f ones; ones[0] = 1.0f; ones[1] = 1.0f;
  v8f s1a = {0.f,0.f,0.f,0.f,0.f,0.f,0.f,0.f};
  v8f s2a = {0.f,0.f,0.f,0.f,0.f,0.f,0.f,0.f};
  v8f s1b = {0.f,0.f,0.f,0.f,0.f,0.f,0.f,0.f};
  v8f s2b = {0.f,0.f,0.f,0.f,0.f,0.f,0.f,0.f};
  const float* base = out + (size_t)(2 * khalf) * FEAT + f0 + nIdx;
  int nMain = n & ~7;
  for (int v0 = 0; v0 < nMain; v0 += 8) {
    const float* p = base + (size_t)v0 * FEAT;
    float e0 = p[0];
    float e1 = p[FEAT];
    float g0 = p[4 * FEAT];
    float g1 = p[5 * FEAT];
    v2f b0; b0[0] = e0;      b0[1] = e1;
    v2f q0; q0[0] = e0 * e0; q0[1] = e1 * e1;
    v2f b1; b1[0] = g0;      b1[1] = g1;
    v2f q1; q1[0] = g0 * g0; q1[1] = g1 * g1;
    s1a = __builtin_amdgcn_wmma_f32_16x16x4_f32(false, ones, false, b0, (short)0, s1a, false, false);
    s2a = __builtin_amdgcn_wmma_f32_16x16x4_f32(false, ones, false, q0, (short)0, s2a, false, false);
    s1b = __builtin_amdgcn_wmma_f32_16x16x4_f32(false, ones, false, b1, (short)0, s1b, false, false);
    s2b = __builtin_amdgcn_wmma_f32_16x16x4_f32(false, ones, false, q1, (short)0, s2b, false, false);
  }
  if (lane < 16) {
    float S1 = s1a[0] + s1b[0];
    float S2 = s2a[0] + s2b[0];
    for (int r = nMain; r < n; ++r) {
      float ev = out[(size_t)r * FEAT + f0 + lane];
      S1 += ev;
      S2 += ev * ev;
    }
    float mean = S1 / (float)n;
    var[f0 + lane] = (S2 - S1 * mean) / (float)(n - 1);  // ddof=1
  }
}

// ---------------- feature selection + RNG ----------------
__global__ void k_select(const float* __restrict__ var, int* lowRank, int* lowIdx,
                         int* highIdx, int* preF, int* randNodes, float* randVals, int n) {
  __shared__ float sv[FEAT];
  __shared__ int isl[FEAT];
  int f = threadIdx.x;
  sv[f] = var[f];
  __syncthreads();
  float vf = sv[f];
  int rank = 0;
  for (int g = 0; g < FEAT; ++g) {
    float vg = sv[g];
    if (vg < vf || (vg == vf && g < f)) rank++;
  }
  bool low = rank < KLOW;
  lowRank[f] = low ? rank : -1;
  isl[f] = low ? 1 : 0;
  if (low) lowIdx[rank] = f;
  __syncthreads();
  if (!low) {
    int nb = 0;
    for (int g = 0; g < f; ++g) nb += isl[g];
    int pos = f - nb;
    highIdx[pos] = f;
    if (pos == 0) preF[0] = f;
  }
  if (f < KLOW) {
    unsigned int h1 = splitmix32(0x1234u + 2u * (unsigned)f);
    unsigned int h2 = splitmix32(0x9876u + 2u * (unsigned)f + 1u);
    randNodes[f] = (int)(h1 % (unsigned int)n);
    randVals[f] = (float)(h2 >> 8) * (1.0f / 16777216.0f);
  }
}

__global__ void k_fixup(float* x2, unsigned char* m2, const int* lowIdx,
                        const int* rn, const float* rv) {
  int i = threadIdx.x;
  if (i < KLOW) {
    size_t off = (size_t)rn[i] * FEAT + lowIdx[i];
    x2[off] = rv[i];
    m2[off] = 1;
  }
}

// ---------------- host ----------------
extern "C" void kernel_launch(void* const* d_in, const int* in_sizes, int n_in,
                              void* d_out, int out_size, void* d_ws, size_t ws_size,
                              hipStream_t stream) {
  const float* x = (const float*)d_in[0];
  const int* ei = (const int*)d_in[1];
  const unsigned char* mask = (const unsigned char*)d_in[2];
  int n = in_sizes[0] / FEAT;
  int e = in_sizes[1] / 2;
  const int* row = ei;
  const int* col = ei + e;
  long long nf = (long long)n * FEAT;

  char* wsp = (char*)d_ws;
  auto alloc = [&](size_t bytes) -> char* {
    char* p = wsp;
    wsp += (bytes + 255) & ~(size_t)255;
    return p;
  };
  int* distA      = (int*)alloc((size_t)n * 4);
  int* distB      = (int*)alloc((size_t)n * 4);
  float* f_n2d    = (float*)alloc((size_t)n * 4);
  float* f_n2d2   = (float*)alloc((size_t)n * 4);
  float* a1       = (float*)alloc((size_t)e * 4);
  float* a_high   = (float*)alloc((size_t)e * 4);
  float* deg      = (float*)alloc((size_t)n * 4);
  int* d12A       = (int*)alloc((size_t)KLOW * n * 4);
  int* d12B       = (int*)alloc((size_t)KLOW * n * 4);
  float* pc       = (float*)alloc((size_t)KLOW * n * 4);
  float* invdl    = (float*)alloc((size_t)KLOW * n * 4);
  float* x2       = (float*)alloc((size_t)nf * 4);
  unsigned char* m2 = (unsigned char*)alloc((size_t)nf);
  float* cur      = (float*)alloc((size_t)nf * 4);
  float* acc      = (float*)alloc((size_t)nf * 4);
  float* var      = (float*)alloc(FEAT * 4);
  int* lowRank    = (int*)alloc(FEAT * 4);
  int* lowIdx     = (int*)alloc(KLOW * 4);
  int* highIdx    = (int*)alloc((FEAT - KLOW) * 4);
  int* preF       = (int*)alloc(4);
  int* randNodes  = (int*)alloc(KLOW * 4);
  float* randVals = (float*)alloc(KLOW * 4);

  const int B = 256;
  int gN  = (n + B - 1) / B;
  int gE  = (e + B - 1) / B;
  int gNF = (int)((nf + B - 1) / B);
  int gKN = (KLOW * n + B - 1) / B;
  int gKE = (int)(((long long)KLOW * e + B - 1) / B);
  // grid-stride SpMM: ~4 edges per wave
  long long waves = ((long long)e + 3) / 4;
  int gEw = (int)((waves + 7) / 8);          // 8 waves per 256-thread block
  if (gEw < 1) gEw = 1;

  // ---- BFS 1 from mask[:,0] ----
  k_bfs_init_maskcol<<<gN, B, 0, stream>>>(distA, mask, 0, n);
  int* dc = distA; int* dn = distB;
  for (int h = 0; h < MAX_HOPS_C; ++h) {
    hipMemcpyAsync(dn, dc, (size_t)n * 4, hipMemcpyDeviceToDevice, stream);
    k_bfs_relax<<<gE, B, 0, stream>>>(dn, dc, row, col, e);
    int* t = dc; dc = dn; dn = t;
  }
  k_bfs_fin<<<gN, B, 0, stream>>>(f_n2d, dc, n);

  // ---- a1 ----
  k_zero<<<gN, B, 0, stream>>>(deg, n);
  k_edge_w<<<gE, B, 0, stream>>>(a1, deg, f_n2d, row, col, e);
  k_norm_edges<<<gE, B, 0, stream>>>(a1, deg, row, e);

  // ---- step1: 20 masked SpMM iterations ----
  k_zero<<<gNF, B, 0, stream>>>(acc, nf);
  k_apply<<<gNF, B, 0, stream>>>(cur, mask, x, acc, nf);
  for (int it = 0; it < NUM_ITER_C; ++it) {
    k_zero<<<gNF, B, 0, stream>>>(acc, nf);
    k_spmm<<<gEw, B, 0, stream>>>(acc, cur, a1, row, col, e);
    k_apply<<<gNF, B, 0, stream>>>(cur, mask, x, acc, nf);
  }

  // ---- per-feature variance (WMMA) + feature split ----
  k_var_wmma<<<FEAT / 16, 32, 0, stream>>>(cur, var, n);
  k_select<<<1, FEAT, 0, stream>>>(var, lowRank, lowIdx, highIdx, preF,
                                   randNodes, randVals, n);

  // ---- x2 / mask2 ----
  hipMemcpyAsync(x2, x, (size_t)nf * 4, hipMemcpyDeviceToDevice, stream);
  hipMemcpyAsync(m2, mask, (size_t)nf, hipMemcpyDeviceToDevice, stream);
  k_fixup<<<1, 32, 0, stream>>>(x2, m2, lowIdx, randNodes, randVals);

  // ---- BFS 2 from mask2[:, pre] ----
  k_bfs_init_precol<<<gN, B, 0, stream>>>(distA, m2, preF, n);
  dc = distA; dn = distB;
  for (int h = 0; h < MAX_HOPS_C; ++h) {
    hipMemcpyAsync(dn, dc, (size_t)n * 4, hipMemcpyDeviceToDevice, stream);
    k_bfs_relax<<<gE, B, 0, stream>>>(dn, dc, row, col, e);
    int* t = dc; dc = dn; dn = t;
  }
  k_bfs_fin<<<gN, B, 0, stream>>>(f_n2d2, dc, n);

  // ---- batched BFS from the 12 random seeds -> pc_low ----
  k_bfs12_init<<<gKN, B, 0, stream>>>(d12A, randNodes, n);
  int* bc = d12A; int* bn = d12B;
  for (int h = 0; h < MAX_HOPS_C; ++h) {
    hipMemcpyAsync(bn, bc, (size_t)KLOW * n * 4, hipMemcpyDeviceToDevice, stream);
    k_bfs12_relax<<<gKE, B, 0, stream>>>(bn, bc, row, col, e, n);
    int* t = bc; bc = bn; bn = t;
  }
  k_pc<<<gKN, B, 0, stream>>>(pc, bc, f_n2d2, n);

  // ---- a_high ----
  k_zero<<<gN, B, 0, stream>>>(deg, n);
  k_edge_w<<<gE, B, 0, stream>>>(a_high, deg, f_n2d2, row, col, e);
  k_norm_edges<<<gE, B, 0, stream>>>(a_high, deg, row, e);

  // ---- deg_low -> inverse (a_low recomputed on the fly in k_spmm2) ----
  k_zero<<<gKN, B, 0, stream>>>(invdl, (long long)KLOW * n);
  k_deglow<<<gKE, B, 0, stream>>>(invdl, pc, row, col, e, n);
  k_inv<<<gKN, B, 0, stream>>>(invdl, (long long)KLOW * n);

  // ---- phase 2: 20 per-column SpMM iterations, final into d_out ----
  k_zero<<<gNF, B, 0, stream>>>(acc, nf);
  k_apply<<<gNF, B, 0, stream>>>(cur, m2, x2, acc, nf);
  float* outf = (float*)d_out;
  for (int it = 0; it < NUM_ITER_C; ++it) {
    k_zero<<<gNF, B, 0, stream>>>(acc, nf);
    k_spmm2<<<gEw, B, 0, stream>>>(acc, cur, a_high, pc, invdl, lowRank,
                                   row, col, e, n);
    float* dst = (it == NUM_ITER_C - 1) ? outf : cur;
    k_apply<<<gNF, B, 0, stream>>>(dst, m2, x2, acc, nf);
  }
}